// HybridEstimatorQNN_65481071403181
// MI455X (gfx1250) — compile-verified
//
#include <hip/hip_runtime.h>
#include <hip/hip_bf16.h>

// ---------------------------------------------------------------------------
// HybridEstimatorQNN on MI455X (gfx1250, wave32).
//
// Roofline: ~50 MB total HBM traffic (8B in + 4B out per row, B=4.19M) ->
// ~2.2 us floor at 23.3 TB/s. Per-row compute (~60 MACs + 12 tanh + 2 sin)
// would make a pure-VALU kernel compute-limited, so all dense-layer MACs run
// on the XDL unit via V_WMMA_F32_16X16X4_F32 (D = W^T x X^T + bias per
// 16-row tile), leaving the VALU for v_sin/v_tanh transcendentals.
//
// IMPORTANT: all ds_swizzle cross-lane moves are hoisted to full-EXEC
// statements (never inside ternaries) — wave32 EXEC must be all-1s for the
// cross-half data exchange and for WMMA.
// ---------------------------------------------------------------------------

typedef __attribute__((ext_vector_type(2))) float v2f;
typedef __attribute__((ext_vector_type(8))) float v8f;

// SWAPX16: lane L <- lane L^16 (ds_swizzle group-of-32: and=0x1f, or=0, xor=0x10)
static __device__ __forceinline__ float swapx16(float v) {
  return __int_as_float(__builtin_amdgcn_ds_swizzle(__float_as_int(v), 0x401F));
}

static __device__ __forceinline__ float fast_tanh(float x) {
#if __has_builtin(__builtin_amdgcn_tanhf)
  return __builtin_amdgcn_tanhf(x);           // V_TANH_F32 (gfx1250 trans op)
#else
  const float e = __expf(2.0f * x);
  return 1.0f - 2.0f * __builtin_amdgcn_rcpf(e + 1.0f);
#endif
}

static __device__ __forceinline__ v8f wmma4(v2f a, v2f b, v8f c) {
  // 8 args: (neg_a, A, neg_b, B, c_mod, C, reuse_a, reuse_b)
  return __builtin_amdgcn_wmma_f32_16x16x4_f32(false, a, false, b,
                                               (short)0, c, false, false);
}

__global__ __launch_bounds__(256) void qnn_mlp_wmma_kernel(
    const float* __restrict__ x,        // [B,2]
    const float* __restrict__ W1,       // [3,8]
    const float* __restrict__ b1,       // [8]
    const float* __restrict__ W2,       // [8,4]
    const float* __restrict__ b2,       // [4]
    const float* __restrict__ W3,       // [4,1]
    const float* __restrict__ b3,       // [1]
    const float* __restrict__ wq,       // [1]
    float* __restrict__ out,            // [B]
    long long nB) {
  const int  lane = (int)(threadIdx.x & 31u);
  const bool lo   = lane < 16;
  const int  m    = lane & 15;

  // ---- uniform scalars (compiler -> s_load) -------------------------------
  const float wphase = wq[0];
  float b1s[8];
#pragma unroll
  for (int j = 0; j < 8; ++j) b1s[j] = b1[j];
  float b2s[4];
#pragma unroll
  for (int j = 0; j < 4; ++j) b2s[j] = b2[j];
  const float w30 = W3[0], w31 = W3[1], w32 = W3[2], w33 = W3[3];
  const float b3s = b3[0];

  // ---- A matrices (weight transposes) in WMMA 16x4 layout -----------------
  // VGPR0: lanes0-15 = K0 row, lanes16-31 = K2 row; VGPR1: K1 / K3.
  v2f a1 = {0.0f, 0.0f};                 // A1[m][k] = W1[k][m], M pad to 16, K pad to 4
  if (m < 8) {
    if (lo) { a1.x = W1[0 * 8 + m]; a1.y = W1[1 * 8 + m]; }
    else    { a1.x = W1[2 * 8 + m]; /* K=3 pad -> 0 */ }
  }
  v2f a2lo = {0.0f, 0.0f}, a2hi = {0.0f, 0.0f};  // A2[m][k] = W2[k][m], K=8 split
  if (m < 4) {
    if (lo) { a2lo.x = W2[0 * 4 + m]; a2lo.y = W2[1 * 4 + m];
              a2hi.x = W2[4 * 4 + m]; a2hi.y = W2[5 * 4 + m]; }
    else    { a2lo.x = W2[2 * 4 + m]; a2lo.y = W2[3 * 4 + m];
              a2hi.x = W2[6 * 4 + m]; a2hi.y = W2[7 * 4 + m]; }
  }

  // ---- C (bias) matrices: VGPR j lanes0-15 = feature j, pad rows -> 0 -----
  v8f c1, c2;
#pragma unroll
  for (int j = 0; j < 8; ++j) c1[j] = lo ? b1s[j] : 0.0f;
#pragma unroll
  for (int j = 0; j < 8; ++j) c2[j] = (lo && j < 4) ? b2s[j] : 0.0f;

  const long long stride = (long long)gridDim.x * (long long)blockDim.x;
  const long long wbase0 =
      (long long)blockIdx.x * blockDim.x + (long long)(threadIdx.x & ~31u);

  for (long long base = wbase0; base < nB; base += stride) {
    const long long row  = base + lane;
    const bool full_tile = (base + 32) <= nB;   // wave-uniform -> scalar branch

    float x0, x1;
    if (full_tile) {                            // unpredicated coalesced b64 load
      float2 v = ((const float2*)x)[row];
      x0 = v.x; x1 = v.y;
    } else {
      x0 = 0.0f; x1 = 0.0f;
      if (row < nB) { float2 v = ((const float2*)x)[row]; x0 = v.x; x1 = v.y; }
    }

    // quantum feature: sin(x0) * sin(x1 + w)   (native v_sin_f32)
    const float q = __sinf(x0) * __sinf(x1 + wphase);

    // ---- layer 1 B tiles (4x16: lanes0-15 = K0/K1, lanes16-31 = K2/K3) ----
    // zs = swapx16({q_lo | x0_hi}) = {x0_hi | q_lo}: serves BOTH tiles.
    const float z   = lo ? q : x0;
    const float zs  = swapx16(z);               // full-EXEC cross-half move
    const float x1s = swapx16(x1);
    v2f bt0, bt1;                               // tile0 = rows +0..15, tile1 = +16..31
    bt0.x = lo ? x0  : zs;                      // {x0_lo | q_lo}
    bt0.y = lo ? x1  : 0.0f;                    // {x1_lo | 0}
    bt1.x = lo ? zs  : q;                       // {x0_hi | q_hi}
    bt1.y = lo ? x1s : 0.0f;                    // {x1_hi | 0}

    v8f h0 = wmma4(a1, bt0, c1);                // D[j][n]: feature j, row n
    v8f h1 = wmma4(a1, bt1, c1);
#pragma unroll
    for (int j = 0; j < 8; ++j) { h0[j] = fast_tanh(h0[j]); h1[j] = fast_tanh(h1[j]); }

    // ---- layer 2: K=8 -> two accumulating WMMAs per tile ------------------
    // All swizzles hoisted (full EXEC); selects become v_cndmask.
    const float s02 = swapx16(h0[2]);
    const float s03 = swapx16(h0[3]);
    const float s06 = swapx16(h0[6]);
    const float s07 = swapx16(h0[7]);
    const float s12 = swapx16(h1[2]);
    const float s13 = swapx16(h1[3]);
    const float s16 = swapx16(h1[6]);
    const float s17 = swapx16(h1[7]);
    v2f blo0, bhi0, blo1, bhi1;
    blo0.x = lo ? h0[0] : s02;
    blo0.y = lo ? h0[1] : s03;
    bhi0.x = lo ? h0[4] : s06;
    bhi0.y = lo ? h0[5] : s07;
    blo1.x = lo ? h1[0] : s12;
    blo1.y = lo ? h1[1] : s13;
    bhi1.x = lo ? h1[4] : s16;
    bhi1.y = lo ? h1[5] : s17;

    v8f g0 = wmma4(a2hi, bhi0, wmma4(a2lo, blo0, c2));
    v8f g1 = wmma4(a2hi, bhi1, wmma4(a2lo, blo1, c2));

    // ---- layer 3 (4 -> 1): 4 FMAs on lanes0-15 of each tile ---------------
    const float r0 = fast_tanh(g0[0]) * w30 + fast_tanh(g0[1]) * w31 +
                     fast_tanh(g0[2]) * w32 + fast_tanh(g0[3]) * w33 + b3s;
    const float r1 = fast_tanh(g1[0]) * w30 + fast_tanh(g1[1]) * w31 +
                     fast_tanh(g1[2]) * w32 + fast_tanh(g1[3]) * w33 + b3s;

    // merge tiles so all 32 lanes store contiguously (coalesced b32 store)
    const float r1s = swapx16(r1);              // full-EXEC
    const float res = lo ? r0 : r1s;
    if (full_tile) {
      __builtin_nontemporal_store(res, &out[row]);
    } else if (row < nB) {
      __builtin_nontemporal_store(res, &out[row]);
    }
  }
}

extern "C" void kernel_launch(void* const* d_in, const int* in_sizes, int n_in,
                              void* d_out, int out_size, void* d_ws, size_t ws_size,
                              hipStream_t stream) {
  const float* x  = (const float*)d_in[0];
  const float* W1 = (const float*)d_in[1];
  const float* b1 = (const float*)d_in[2];
  const float* W2 = (const float*)d_in[3];
  const float* b2 = (const float*)d_in[4];
  const float* W3 = (const float*)d_in[5];
  const float* b3 = (const float*)d_in[6];
  const float* wq = (const float*)d_in[7];
  float* out = (float*)d_out;

  const long long nB = (long long)(in_sizes[0] / 2);  // inputs are [B,2]

  // Grid-stride: 2048 blocks x 256 threads = 16K waves; each wave handles
  // 32 rows/iter (~8 iters at B=4.19M), amortizing per-thread weight setup.
  const int threads = 256;
  int blocks = 2048;
  const long long waves_needed = (nB + 31) / 32;
  const long long waves_per_grid = (long long)blocks * (threads / 32);
  if (waves_per_grid > waves_needed && waves_needed > 0) {
    blocks = (int)((waves_needed + (threads / 32) - 1) / (threads / 32));
  }

  hipLaunchKernelGGL(qnn_mlp_wmma_kernel, dim3(blocks), dim3(threads), 0, stream,
                     x, W1, b1, W2, b2, W3, b3, wq, out, nB);
}